// DecoderRNN_39384850104452
// MI455X (gfx1250) — compile-verified
//
#include <hip/hip_runtime.h>
#include <math.h>

// Problem constants (match reference)
#define B_   32
#define T_   64
#define H_   1024
#define G_   3072      // 3*H
#define V_   32000
#define M_   (B_ * T_) // 2048
#define SOS_ 1

typedef __attribute__((ext_vector_type(2))) float v2f;
typedef __attribute__((ext_vector_type(8))) float v8f;

// ---------------------------------------------------------------------------
// Embedding gather + ReLU:  x[b*T+t, :] = relu(emb[token(b,t), :])
// ---------------------------------------------------------------------------
__global__ __launch_bounds__(256) void embed_relu_kernel(
    const int* __restrict__ target, const float* __restrict__ emb,
    float* __restrict__ x)
{
  const int row = blockIdx.x;          // b*T + t
  const int b = row / T_, t = row % T_;
  const int tok = (t == 0) ? SOS_ : target[b * T_ + (t - 1)];
  const float* e = emb + (size_t)tok * H_;
  float* xr = x + (size_t)row * H_;
  for (int j = threadIdx.x; j < H_; j += 256)
    xr[j] = fmaxf(e[j], 0.0f);
}

// ---------------------------------------------------------------------------
// fp32 WMMA GEMM:  C[M,N] = A[M,K] * W[N,K]^T + bias[N]
// Block = 256 threads (8 waves). Block tile: 32(M) x 512(N).
// Each wave owns a 32x64 strip -> 8 accumulators (2 M-tiles x 4 N-tiles).
// LDS tiles are stored FRAGMENT-MAJOR: within each 16-wide K group the
// elements are permuted  pos(k) = (k&1) + ((k>>2)<<1) + ((k&2) ? 8 : 0)
// so a lane's 8 fragment floats (lanes 0-15: k∈{0,1,4,5,8,9,12,13};
// lanes 16-31: k∈{2,3,6,7,10,11,14,15}) are contiguous -> 2x ds_load_b128
// per fragment, letting 32 v_wmma_f32_16x16x4_f32 issue per K-step with
// minimal DS waits. Row stride padded to 20 floats (80B, 16B aligned,
// bank-conflict free across 16 lanes).
// Requires: M % 32 == 0, N % 64 == 0, K % 16 == 0.
// ---------------------------------------------------------------------------
#define KSTEP  16
#define BLK_M  32
#define BLK_N  512
#define WPAD   20

__device__ __forceinline__ void stage_frag_row(float* dst, float4 v, int kq) {
  const int p = kq >> 1;               // kq in {0,4,8,12} -> p in {0,2,4,6}
  dst[p]     = v.x;  dst[p + 1] = v.y; // k, k+1  -> first 8-group
  dst[8 + p] = v.z;  dst[9 + p] = v.w; // k+2,k+3 -> second 8-group
}

__global__ __launch_bounds__(256) void gemm_wmma_f32_kernel(
    const float* __restrict__ A, const float* __restrict__ W,
    const float* __restrict__ bias, float* __restrict__ C,
    int M, int N, int K)
{
  __shared__ __align__(16) float As[BLK_M][WPAD];
  __shared__ __align__(16) float Ws[BLK_N][WPAD];

  const int tid  = threadIdx.x;
  const int lane = tid & 31;
  const int wv   = tid >> 5;                 // wave 0..7
  const int m0   = blockIdx.x * BLK_M;
  const int nB   = blockIdx.y * BLK_N;       // block N base
  const int nW   = nB + wv * 64;             // this wave's N base
  const int half = lane >> 4;                // 0: lanes 0-15, 1: lanes 16-31
  const int lm   = lane & 15;
  const int hb   = half << 3;                // fragment float offset (0 or 8)

  v8f acc[2][4];
  #pragma unroll
  for (int i = 0; i < 2; ++i)
    #pragma unroll
    for (int j = 0; j < 4; ++j)
      acc[i][j] = (v8f){};

  for (int k0 = 0; k0 < K; k0 += KSTEP) {
    __syncthreads();
    // --- stage A tile: 32 rows x 16 k (128 threads x float4) ---
    if (tid < 128) {
      const int r = tid >> 2, kq = (tid & 3) << 2;
      const float4 v = *(const float4*)(A + (size_t)(m0 + r) * K + k0 + kq);
      stage_frag_row(&As[r][0], v, kq);
    }
    // --- stage W tile: BLK_N rows x 16 k; prefetch next K tile ---
    {
      const int r0 = tid >> 2, kq = (tid & 3) << 2;
      #pragma unroll
      for (int rr = r0; rr < BLK_N; rr += 64) {
        const int gn = nB + rr;
        if (gn < N) {
          const float* src = W + (size_t)gn * K + k0 + kq;
          const float4 v = *(const float4*)src;
          stage_frag_row(&Ws[rr][0], v, kq);
          if (k0 + KSTEP < K)
            __builtin_prefetch((const void*)(src + KSTEP), 0, 0);
        }
      }
    }
    __syncthreads();
    if (nW >= N) continue;                   // guard wave (N multiple of 64)

    // --- load all fragments for this K-step (ds_load_b128 x12), then MMA ---
    float4 af[2][2];
    #pragma unroll
    for (int i = 0; i < 2; ++i) {
      const float4* pa = (const float4*)(&As[(i << 4) + lm][hb]);
      af[i][0] = pa[0]; af[i][1] = pa[1];
    }
    float4 bf[4][2];
    #pragma unroll
    for (int j = 0; j < 4; ++j) {
      const float4* pb = (const float4*)(&Ws[(wv << 6) + (j << 4) + lm][hb]);
      bf[j][0] = pb[0]; bf[j][1] = pb[1];
    }
    // 32 back-to-back v_wmma_f32_16x16x4_f32 (K = 4 x 4 steps)
    #pragma unroll
    for (int q = 0; q < 4; ++q) {
      #pragma unroll
      for (int i = 0; i < 2; ++i) {
        v2f a;
        a.x = (q & 1) ? af[i][q >> 1].z : af[i][q >> 1].x;
        a.y = (q & 1) ? af[i][q >> 1].w : af[i][q >> 1].y;
        #pragma unroll
        for (int j = 0; j < 4; ++j) {
          v2f b;
          b.x = (q & 1) ? bf[j][q >> 1].z : bf[j][q >> 1].x;
          b.y = (q & 1) ? bf[j][q >> 1].w : bf[j][q >> 1].y;
          acc[i][j] = __builtin_amdgcn_wmma_f32_16x16x4_f32(
              false, a, false, b, (short)0, acc[i][j], false, false);
        }
      }
    }
  }

  // --- epilogue: C layout — VGPR r holds rows (base+r) / (base+r+8) ---
  if (nW < N) {
    #pragma unroll
    for (int j = 0; j < 4; ++j) {
      const int col = nW + (j << 4) + lm;
      const float bv = bias[col];
      #pragma unroll
      for (int i = 0; i < 2; ++i) {
        #pragma unroll
        for (int r = 0; r < 8; ++r) {
          const int row = m0 + (i << 4) + r + (half << 3);
          C[(size_t)row * N + col] = acc[i][j][r] + bv;
        }
      }
    }
  }
}

// ---------------------------------------------------------------------------
// GRU elementwise update for one time step t (after hg GEMM)
// ---------------------------------------------------------------------------
__global__ __launch_bounds__(256) void gru_step_kernel(
    const float* __restrict__ xg, const float* __restrict__ hg,
    float* __restrict__ h, float* __restrict__ hs, int t)
{
  const int i = blockIdx.x * 256 + threadIdx.x;   // 0 .. B*H-1
  const int b = i >> 10;                          // H = 1024
  const int j = i & (H_ - 1);
  const float* xrow = xg + (size_t)(b * T_ + t) * G_;
  const float* hrow = hg + (size_t)b * G_;
  const float xr = xrow[j], xz = xrow[H_ + j], xn = xrow[2 * H_ + j];
  const float hr = hrow[j], hz = hrow[H_ + j], hn = hrow[2 * H_ + j];
  const float r = 1.0f / (1.0f + expf(-(xr + hr)));
  const float z = 1.0f / (1.0f + expf(-(xz + hz)));
  const float n = tanhf(xn + r * hn);
  const float hp = h[i];
  const float hnew = (1.0f - z) * n + z * hp;
  h[i] = hnew;
  hs[(size_t)(b * T_ + t) * H_ + j] = hnew;
}

// ---------------------------------------------------------------------------
// In-place log_softmax per row of V_ elements. One 256-thread block per row.
// ---------------------------------------------------------------------------
__global__ __launch_bounds__(256) void logsoftmax_rows_kernel(float* __restrict__ out)
{
  float* row = out + (size_t)blockIdx.x * V_;
  float m = -INFINITY, s = 0.0f;
  for (int v = threadIdx.x; v < V_; v += 256) {
    const float xv = row[v];
    const float nm = fmaxf(m, xv);
    s = s * expf(m - nm) + expf(xv - nm);
    m = nm;
  }
  #pragma unroll
  for (int off = 16; off > 0; off >>= 1) {
    const float m2 = __shfl_xor(m, off, 32);
    const float s2 = __shfl_xor(s, off, 32);
    const float nm = fmaxf(m, m2);
    s = s * expf(m - nm) + s2 * expf(m2 - nm);
    m = nm;
  }
  __shared__ float mm[8], ss[8];
  __shared__ float fsub;
  if ((threadIdx.x & 31) == 0) { mm[threadIdx.x >> 5] = m; ss[threadIdx.x >> 5] = s; }
  __syncthreads();
  if (threadIdx.x == 0) {
    float M = mm[0], S = ss[0];
    #pragma unroll
    for (int i = 1; i < 8; ++i) {
      const float nm = fmaxf(M, mm[i]);
      S = S * expf(M - nm) + ss[i] * expf(mm[i] - nm);
      M = nm;
    }
    fsub = M + logf(S);
  }
  __syncthreads();
  const float sub = fsub;
  for (int v = threadIdx.x; v < V_; v += 256)
    row[v] = row[v] - sub;
}

// ---------------------------------------------------------------------------
// Host-side orchestration (graph-capture safe)
// ---------------------------------------------------------------------------
extern "C" void kernel_launch(void* const* d_in, const int* in_sizes, int n_in,
                              void* d_out, int out_size, void* d_ws, size_t ws_size,
                              hipStream_t stream) {
  (void)in_sizes; (void)n_in; (void)out_size; (void)ws_size;

  const float* enc_h  = (const float*)d_in[1];   // [1,B,H]
  const int*   target = (const int*)d_in[2];     // [B,T]
  const float* emb    = (const float*)d_in[3];   // [V,H]
  const float* W_ih   = (const float*)d_in[4];   // [3H,H]
  const float* W_hh   = (const float*)d_in[5];   // [3H,H]
  const float* b_ih   = (const float*)d_in[6];   // [3H]
  const float* b_hh   = (const float*)d_in[7];   // [3H]
  const float* out_W  = (const float*)d_in[8];   // [V,H]
  const float* out_b  = (const float*)d_in[9];   // [V]
  float* out = (float*)d_out;                    // [B,T,V] logprobs ++ [1,B,H]

  float* ws = (float*)d_ws;
  float* x  = ws;                                  // 2048*1024
  float* xg = x  + (size_t)M_ * H_;                // 2048*3072
  float* hs = xg + (size_t)M_ * G_;                // 2048*1024
  float* h  = hs + (size_t)M_ * H_;                // 32*1024
  float* hg = h  + (size_t)B_ * H_;                // 32*3072

  // 1) x = relu(emb[tokens])
  embed_relu_kernel<<<M_, 256, 0, stream>>>(target, emb, x);

  // 2) xg = x @ W_ih^T + b_ih      (2048 x 3072 x 1024)
  {
    dim3 grid(M_ / BLK_M, G_ / BLK_N);   // 64 x 6
    gemm_wmma_f32_kernel<<<grid, 256, 0, stream>>>(x, W_ih, b_ih, xg, M_, G_, H_);
  }

  // 3) h = encoder_hidden[0]
  hipMemcpyAsync(h, enc_h, (size_t)B_ * H_ * sizeof(float),
                 hipMemcpyDeviceToDevice, stream);

  // 4) sequential GRU scan (dependency chain -> per-step launches in the graph)
  for (int t = 0; t < T_; ++t) {
    dim3 grid(B_ / BLK_M, G_ / BLK_N);   // 1 x 6
    gemm_wmma_f32_kernel<<<grid, 256, 0, stream>>>(h, W_hh, b_hh, hg, B_, G_, H_);
    gru_step_kernel<<<(B_ * H_) / 256, 256, 0, stream>>>(xg, hg, h, hs, t);
  }

  // 5) logits = hs @ out_W^T + out_b  -> straight into d_out (2048 x 32000 x 1024)
  {
    dim3 grid(M_ / BLK_M, (V_ + BLK_N - 1) / BLK_N);   // 64 x 63
    gemm_wmma_f32_kernel<<<grid, 256, 0, stream>>>(hs, out_W, out_b, out, M_, V_, H_);
  }

  // 6) in-place log_softmax over each of the 2048 rows
  logsoftmax_rows_kernel<<<M_, 256, 0, stream>>>(out);

  // 7) append h_last ([1,B,H]) after the logprobs
  hipMemcpyAsync(out + (size_t)M_ * V_, h, (size_t)B_ * H_ * sizeof(float),
                 hipMemcpyDeviceToDevice, stream);
}